// SphericalHarmonicsNoSphericart_43499428774526
// MI455X (gfx1250) — compile-verified
//
#include <hip/hip_runtime.h>

#ifndef __has_builtin
#define __has_builtin(x) 0
#endif

#define LMAX 8
#define NSH  81      // (LMAX+1)^2
#define NQ   45      // (LMAX+1)(LMAX+2)/2
#define PTS  128     // points (= threads) per block; 128*81*4 = 41472 B LDS tile

// gfx12-style CPol: TH[2:0], SCOPE[4:3]. TH_NT = 1 -> non-temporal streaming store.
#define CPOL_NT 1

typedef int v4i_t __attribute__((ext_vector_type(4)));
typedef __attribute__((address_space(1))) v4i_t* gv4_ptr;   // global int4*
typedef __attribute__((address_space(3))) v4i_t* lv4_ptr;   // LDS int4*

__global__ __launch_bounds__(PTS) void sh_l8_kernel(
    const float* __restrict__ xyz, const float* __restrict__ F,
    float* __restrict__ out, int npts) {
  __shared__ float tile[PTS * NSH];

  const int tid = threadIdx.x;
  const int p0  = blockIdx.x * PTS;
  const int p   = p0 + tid;
  const int nvalid = min(PTS, npts - p0);

  // 45 normalization constants: uniform address -> scalar (SMEM) loads
  float Fr[NQ];
  #pragma unroll
  for (int i = 0; i < NQ; ++i) Fr[i] = F[i];

  if (tid < nvalid) {
    float x = xyz[3 * p + 0];
    float y = xyz[3 * p + 1];
    float z = xyz[3 * p + 2];
    const float rinv = rsqrtf(x * x + y * y + z * z);
    x *= rinv; y *= rinv; z *= rinv;

    // Associated-Legendre style recurrence, fully unrolled, idx(l,m)=l(l+1)/2+m
    float Q[NQ];
    Q[0] = 1.0f;
    #pragma unroll
    for (int l = 1; l <= LMAX; ++l) {
      const int row   = l * (l + 1) / 2;
      const int prow  = (l - 1) * l / 2;
      const int pprow = (l >= 2) ? (l - 2) * (l - 1) / 2 : 0;
      Q[row + l]     = -(float)(2 * l - 1) * Q[prow + l - 1];
      Q[row + l - 1] = -z * Q[row + l];
      #pragma unroll
      for (int m = 0; m <= l - 2; ++m) {
        Q[row + m] = ((float)(2 * l - 1) * z * Q[prow + m]
                      - (float)(l + m - 1) * Q[pprow + m])
                     * (1.0f / (float)(l - m));
      }
    }

    // sin/cos multiple-angle recurrence
    float s[LMAX + 1], c[LMAX + 1];
    s[0] = 0.0f; c[0] = 1.0f;
    #pragma unroll
    for (int m = 1; m <= LMAX; ++m) {
      s[m] = x * s[m - 1] + y * c[m - 1];
      c[m] = x * c[m - 1] - y * s[m - 1];
    }

    const float INVSQRT2 = 0.70710678118654752440f;
    float* rowo = &tile[tid * NSH];   // stride 81 dwords -> bank-conflict-free
    rowo[0] = Fr[0] * Q[0] * INVSQRT2;
    #pragma unroll
    for (int l = 1; l <= LMAX; ++l) {
      const int b = l * (l + 1) / 2;
      const int o = l * l + l;
      #pragma unroll
      for (int m = 1; m <= l; ++m) rowo[o - m] = Fr[b + m] * Q[b + m] * s[m];
      rowo[o] = Fr[b] * Q[b] * INVSQRT2;
      #pragma unroll
      for (int m = 1; m <= l; ++m) rowo[o + m] = Fr[b + m] * Q[b + m] * c[m];
    }
  }

  __syncthreads();

  // Stream the whole LDS tile to global with coalesced 16B chunks per lane,
  // via the CDNA5 async LDS->global data mover (ASYNCcnt-tracked), marked
  // non-temporal so the 324MB output stream doesn't thrash L2.
  // nvalid*324 bytes is always a multiple of 16 (nvalid is 128 or 64 here).
  const int nchunks = (nvalid * NSH * 4) >> 4;
  char* gbase = (char*)out + (size_t)p0 * (NSH * 4);
  for (int ch = tid; ch < nchunks; ch += PTS) {
    __builtin_amdgcn_global_store_async_from_lds_b128(
        (gv4_ptr)(gbase + (size_t)ch * 16),
        (lv4_ptr)((char*)tile + ch * 16),
        0, CPOL_NT);
  }
#if __has_builtin(__builtin_amdgcn_s_wait_asynccnt)
  __builtin_amdgcn_s_wait_asynccnt(0);
#endif
}

extern "C" void kernel_launch(void* const* d_in, const int* in_sizes, int n_in,
                              void* d_out, int out_size, void* d_ws, size_t ws_size,
                              hipStream_t stream) {
  const float* xyz = (const float*)d_in[0];
  const float* F   = (const float*)d_in[1];
  float* out       = (float*)d_out;
  const int npts   = in_sizes[0] / 3;

  const int nblocks = (npts + PTS - 1) / PTS;
  sh_l8_kernel<<<dim3(nblocks), dim3(PTS), 0, stream>>>(xyz, F, out, npts);
}